// SAMBlock_27934467293444
// MI455X (gfx1250) — compile-verified
//
#include <hip/hip_runtime.h>

// fp32 WMMA fragment types (wave32):
//  A: 16x4 f32 -> 2 VGPRs/lane, B: 4x16 f32 -> 2 VGPRs/lane, C/D: 16x16 f32 -> 8 VGPRs/lane
typedef __attribute__((ext_vector_type(2))) float v2f;
typedef __attribute__((ext_vector_type(8))) float v8f;
// TDM descriptor operand types (6-arg builtin form: g0, g1, g2, g3, g4, cpol)
typedef __attribute__((ext_vector_type(4))) unsigned int u32x4;
typedef __attribute__((ext_vector_type(8))) int i32x8;
typedef __attribute__((ext_vector_type(4))) int i32x4;

#define WMMA_F32X4(acc, a, b) \
  __builtin_amdgcn_wmma_f32_16x16x4_f32(false, (a), false, (b), (short)0, (acc), false, false)

constexpr int BATCH = 2048;
constexpr int NF    = 40;   // num fields
constexpr int DD    = 64;   // embedding dim
constexpr int NP    = 48;   // padded N (3 tiles of 16); padded rows stay zero
constexpr int SF    = 68;   // LDS stride (floats): 16B-aligned rows, conflict-free (68 mod 64 = 4)
constexpr int SS    = 52;   // LDS stride for 48-wide S buffer (52 = 4*13, gcd(13,16)=1)
constexpr int NLAYER = 3;

__global__ __launch_bounds__(128) void sam3e_kernel(
    const float* __restrict__ Fin,  // (B, N, D)
    const float* __restrict__ Kw,   // (L, D, D)
    const float* __restrict__ Qw,   // (L, D, D)
    const float* __restrict__ wvec, // (N, 1)
    float* __restrict__ outp)       // (B, D)
{
  __shared__ float Fs[NP * SF];   // current F (padded 48 x 64, row stride 68)
  __shared__ float Ts[NP * SF];   // scratch: holds KF, then S@F
  __shared__ float Ss[NP * SS];   // similarity S (padded 48 x 48)
  __shared__ float Ws[DD * SF];   // layer weight: Kw, then reused for Qw

  const int b    = blockIdx.x;
  const int tid  = threadIdx.x;
  const int lane = tid & 31;
  // Wave id, forced scalar-uniform so all WMMA control flow is scalar branches
  // (ISA 7.12: EXEC must be all 1's for WMMA).
  const int wv   = __builtin_amdgcn_readfirstlane(tid) >> 5;   // 0..3
  const int lm   = lane & 15;          // fragment row (A) / col (B,C,D)
  const int lk   = (lane >> 4) * 2;    // K sub-offset for A/B fragments
  const int cm   = (lane >> 4) * 8;    // M sub-offset for C/D fragments
  const int nb   = wv * 16 + lm;       // this wave's output-column lane index

  // ---- Load F(b) into LDS via the Tensor Data Mover (one DMA for the 40x64 tile).
  // TDM LDS padding: 64 dwords then +4 dwords pad => LDS row stride 68 floats == SF.
  if (wv == 0) {
    const unsigned long long ga =
        (unsigned long long)(uintptr_t)(Fin + (size_t)b * NF * DD);
    const unsigned int la = (unsigned int)(uintptr_t)(&Fs[0]);  // LDS byte offset
    u32x4 g0;
    g0[0] = 1u;                                     // count=1, user descriptor
    g0[1] = la;                                     // lds_addr
    g0[2] = (unsigned int)(ga & 0xFFFFFFFFu);       // global_addr[31:0]
    g0[3] = (unsigned int)((ga >> 32) & 0x01FFFFFFu) | (2u << 30);  // addr[56:32] | type=2
    i32x8 g1;
    g1[0] = (int)((2u << 16)        // data_size = 4 bytes
                | (1u << 20)        // pad_enable
                | (5u << 22)        // pad_interval: 2^(5+1) = 64 dwords
                | (3u << 25));      // pad_amount: 3+1 = 4 dwords
    g1[1] = (int)(64u << 16);       // tensor_dim0 = 64 (bits 79:48)
    g1[2] = (int)(40u << 16);       // tensor_dim1 = 40 (bits 111:80)
    g1[3] = (int)(64u << 16);       // tile_dim0 = 64 (bits 127:112)
    g1[4] = (int)40u;               // tile_dim1 = 40, tile_dim2 = 0
    g1[5] = 64;                     // tensor_dim0_stride = 64 elements
    g1[6] = 0;
    g1[7] = 0;
    const i32x4 gz4 = {0, 0, 0, 0};            // groups 2/3 unused (2D tensor)
    const i32x8 gz8 = {0, 0, 0, 0, 0, 0, 0, 0};
    __builtin_amdgcn_tensor_load_to_lds(g0, g1, gz4, gz4, gz8, 0);
  } else {
    // Zero the padding rows 40..47 (cols 0..63) while the DMA runs.
    for (int i = tid - 32; i < 8 * 16; i += 96) {         // 128 float4 slots
      const int r = NF + (i >> 4), c4 = (i & 15) << 2;
      float4 z = {0.f, 0.f, 0.f, 0.f};
      *reinterpret_cast<float4*>(&Fs[r * SF + c4]) = z;
    }
  }
  if (wv == 0) __builtin_amdgcn_s_wait_tensorcnt(0);
  __syncthreads();

  for (int l = 0; l < NLAYER; ++l) {
    const float* Kl = Kw + (size_t)l * DD * DD;
    const float* Ql = Qw + (size_t)l * DD * DD;

    // Ws <- Kw[l]  (float4 / b128 path; LDS rows are 16B aligned with SF=68)
    for (int i = tid; i < DD * 16; i += 128) {
      const float4 v = reinterpret_cast<const float4*>(Kl)[i];
      *reinterpret_cast<float4*>(&Ws[(i >> 4) * SF + ((i & 15) << 2)]) = v;
    }
    __syncthreads();

    // ---- Stage 1: KF = F @ Kw^T -> Ts. Wave owns column-tile nt=wv, rows mt=0..2.
    // B fragment shared across the 3 M-tiles; 3 independent WMMA accumulate chains.
    {
      v8f acc0 = {}, acc1 = {}, acc2 = {};
      #pragma unroll
      for (int kk = 0; kk < 16; ++kk) {
        const int kg = kk * 4 + lk;
        v2f bb;  // B[k=d][n=e] = Kw[e][d] -> row read of Ws
        bb.x = Ws[nb * SF + kg];     bb.y = Ws[nb * SF + kg + 1];
        v2f a0, a1, a2;
        a0.x = Fs[lm * SF + kg];        a0.y = Fs[lm * SF + kg + 1];
        a1.x = Fs[(16 + lm) * SF + kg]; a1.y = Fs[(16 + lm) * SF + kg + 1];
        a2.x = Fs[(32 + lm) * SF + kg]; a2.y = Fs[(32 + lm) * SF + kg + 1];
        acc0 = WMMA_F32X4(acc0, a0, bb);
        acc1 = WMMA_F32X4(acc1, a1, bb);
        acc2 = WMMA_F32X4(acc2, a2, bb);
      }
      #pragma unroll
      for (int r = 0; r < 8; ++r) {
        Ts[(cm + r) * SF + nb]      = acc0[r];
        Ts[(16 + cm + r) * SF + nb] = acc1[r];
        Ts[(32 + cm + r) * SF + nb] = acc2[r];
      }
    }
    __syncthreads();

    // ---- Stage 2: S = F @ KF^T -> Ss (3x3 tiles; waves 0..2).
    // Wave 3 overlaps the Qw[l] -> Ws fill (stage 4 reads Ws after the next syncs).
    if (wv == 3) {
      for (int i = lane; i < DD * 16; i += 32) {
        const float4 v = reinterpret_cast<const float4*>(Ql)[i];
        *reinterpret_cast<float4*>(&Ws[(i >> 4) * SF + ((i & 15) << 2)]) = v;
      }
    } else {
      v8f acc0 = {}, acc1 = {}, acc2 = {};
      #pragma unroll
      for (int kk = 0; kk < 16; ++kk) {
        const int kg = kk * 4 + lk;
        v2f bb;  // B[k=d][n=m] = KF[m][d] -> row read of Ts
        bb.x = Ts[nb * SF + kg];     bb.y = Ts[nb * SF + kg + 1];
        v2f a0, a1, a2;
        a0.x = Fs[lm * SF + kg];        a0.y = Fs[lm * SF + kg + 1];
        a1.x = Fs[(16 + lm) * SF + kg]; a1.y = Fs[(16 + lm) * SF + kg + 1];
        a2.x = Fs[(32 + lm) * SF + kg]; a2.y = Fs[(32 + lm) * SF + kg + 1];
        acc0 = WMMA_F32X4(acc0, a0, bb);
        acc1 = WMMA_F32X4(acc1, a1, bb);
        acc2 = WMMA_F32X4(acc2, a2, bb);
      }
      #pragma unroll
      for (int r = 0; r < 8; ++r) {
        Ss[(cm + r) * SS + nb]      = acc0[r];
        Ss[(16 + cm + r) * SS + nb] = acc1[r];
        Ss[(32 + cm + r) * SS + nb] = acc2[r];
      }
    }
    __syncthreads();

    // ---- Stage 3: T = S @ F -> Ts (KF dead). nt=wv, mt=0..2, K=48.
    {
      v8f acc0 = {}, acc1 = {}, acc2 = {};
      #pragma unroll
      for (int kk = 0; kk < 12; ++kk) {
        const int kg = kk * 4 + lk;
        v2f bb;  // B[k=m][n=d] = F[m][d] -> direct read of Fs, shared across M-tiles
        bb.x = Fs[kg * SF + nb];     bb.y = Fs[(kg + 1) * SF + nb];
        v2f a0, a1, a2;
        a0.x = Ss[lm * SS + kg];        a0.y = Ss[lm * SS + kg + 1];
        a1.x = Ss[(16 + lm) * SS + kg]; a1.y = Ss[(16 + lm) * SS + kg + 1];
        a2.x = Ss[(32 + lm) * SS + kg]; a2.y = Ss[(32 + lm) * SS + kg + 1];
        acc0 = WMMA_F32X4(acc0, a0, bb);
        acc1 = WMMA_F32X4(acc1, a1, bb);
        acc2 = WMMA_F32X4(acc2, a2, bb);
      }
      #pragma unroll
      for (int r = 0; r < 8; ++r) {
        Ts[(cm + r) * SF + nb]      = acc0[r];
        Ts[(16 + cm + r) * SF + nb] = acc1[r];
        Ts[(32 + cm + r) * SF + nb] = acc2[r];
      }
    }
    __syncthreads();

    // ---- Stage 4: out = F .* T + F @ Qw^T, held in registers, then Fs <- out.
    float od0[8], od1[8], od2[8];
    {
      v8f acc0 = {}, acc1 = {}, acc2 = {};
      #pragma unroll
      for (int kk = 0; kk < 16; ++kk) {
        const int kg = kk * 4 + lk;
        v2f bb;  // B[k=d][n=e] = Qw[e][d]
        bb.x = Ws[nb * SF + kg];     bb.y = Ws[nb * SF + kg + 1];
        v2f a0, a1, a2;
        a0.x = Fs[lm * SF + kg];        a0.y = Fs[lm * SF + kg + 1];
        a1.x = Fs[(16 + lm) * SF + kg]; a1.y = Fs[(16 + lm) * SF + kg + 1];
        a2.x = Fs[(32 + lm) * SF + kg]; a2.y = Fs[(32 + lm) * SF + kg + 1];
        acc0 = WMMA_F32X4(acc0, a0, bb);
        acc1 = WMMA_F32X4(acc1, a1, bb);
        acc2 = WMMA_F32X4(acc2, a2, bb);
      }
      #pragma unroll
      for (int r = 0; r < 8; ++r) {
        const int m0 = cm + r, m1 = 16 + cm + r, m2 = 32 + cm + r;
        od0[r] = Fs[m0 * SF + nb] * Ts[m0 * SF + nb] + acc0[r];
        od1[r] = Fs[m1 * SF + nb] * Ts[m1 * SF + nb] + acc1[r];
        od2[r] = Fs[m2 * SF + nb] * Ts[m2 * SF + nb] + acc2[r];
      }
    }
    __syncthreads();   // everyone done reading old F
    #pragma unroll
    for (int r = 0; r < 8; ++r) {
      Fs[(cm + r) * SF + nb]      = od0[r];
      Fs[(16 + cm + r) * SF + nb] = od1[r];
      Fs[(32 + cm + r) * SF + nb] = od2[r];
    }
    __syncthreads();
  }

  // ---- Weighted pooling: out[b,d] = sum_n F[n,d] * w[n]
  if (tid < DD) {
    float s = 0.0f;
    for (int n = 0; n < NF; ++n)
      s += Fs[n * SF + tid] * wvec[n];
    outp[(size_t)b * DD + tid] = s;
  }
}

extern "C" void kernel_launch(void* const* d_in, const int* in_sizes, int n_in,
                              void* d_out, int out_size, void* d_ws, size_t ws_size,
                              hipStream_t stream) {
  const float* F = (const float*)d_in[0];  // (2048, 40, 64)
  const float* K = (const float*)d_in[1];  // (3, 64, 64)
  const float* Q = (const float*)d_in[2];  // (3, 64, 64)
  const float* w = (const float*)d_in[3];  // (40, 1)
  float* out = (float*)d_out;              // (2048, 64)
  (void)in_sizes; (void)n_in; (void)out_size; (void)d_ws; (void)ws_size;

  sam3e_kernel<<<dim3(BATCH), dim3(128), 0, stream>>>(F, K, Q, w, out);
}